// Flow_197568495730
// MI455X (gfx1250) — compile-verified
//
#include <hip/hip_runtime.h>

typedef __attribute__((ext_vector_type(2))) float v2f;
typedef __attribute__((ext_vector_type(8))) float v8f;

#define R_DIM 450
#define C_DIM 480
#define PLANE (R_DIM * C_DIM)   // 216000
#define B_DIM 8
#define C_IN  8
#define F_DIM 4
#define K_DIM 32                // F_DIM * C_IN
#define O_DIM 16

// One wave32 handles one 16-pixel tile (N) x all 16 output channels (M) x all 8
// batches, accumulating K=32 via eight V_WMMA_F32_16X16X4_F32 steps per batch.
// A (combinator) layout, 32-bit 16x4: lane l<16 holds A[M=l][K=0,1] in v0,v1;
// lane l>=16 holds A[M=l-16][K=2,3]. B (warped) 4x16 is symmetric:
// lane l<16 holds B[K=0,1][N=l]; lane l>=16 holds B[K=2,3][N=l-16].
// D 16x16: vgpr v, lane l -> M = v + 8*(l>>4), N = l&15.
__global__ __launch_bounds__(128) void flow_combine_wmma(
    const float* __restrict__ x,      // (B, C_IN, R, C)
    const float* __restrict__ flow,   // (F, R, C, 2)
    const float* __restrict__ comb,   // (O, K)
    float* __restrict__ out)          // (B, O, R, C)
{
  const int lane = threadIdx.x & 31;
  const int wave = threadIdx.x >> 5;
  const int tile = blockIdx.x * 4 + wave;   // 13500 tiles, exact
  const int half = lane >> 4;               // 0: K={0,1}(mod4), 1: K={2,3}(mod4)
  const int l16  = lane & 15;

  const int p  = tile * 16 + l16;           // flat pixel index (never straddles a row)
  const int r  = p / C_DIM;
  const int cc = p % C_DIM;

  // ---- A tiles: combinator, 8 x (16x4) f32, shared by all waves ----
  v2f a[8];
  #pragma unroll
  for (int kk = 0; kk < 8; ++kk) {
    const int kb = kk * 4 + half * 2;       // even -> 8B aligned
    a[kk] = *(const v2f*)(comb + l16 * K_DIM + kb);
  }

  // ---- per-pixel inverse-flow gather params for each of the 4 flows ----
  int   off[F_DIM][4];
  float wgt[F_DIM][4];
  #pragma unroll
  for (int f = 0; f < F_DIM; ++f) {
    const float* fp = flow + ((size_t)(f * R_DIM + r) * C_DIM + cc) * 2;
    const float ix = (float)r  - fp[0];
    const float iy = (float)cc - fp[1];
    const int   bx = (int)rintf(ix);        // jnp.round: half-to-even == RNE
    const int   by = (int)rintf(iy);
    const float s  = ix - floorf(ix);
    const float t  = iy - floorf(iy);
    const int cx0 = min(max(bx,     0), R_DIM - 1);
    const int cx1 = min(max(bx + 1, 0), R_DIM - 1);
    const int cy0 = min(max(by,     0), C_DIM - 1);
    const int cy1 = min(max(by + 1, 0), C_DIM - 1);
    off[f][0] = cx0 * C_DIM + cy0;  wgt[f][0] = s * t;
    off[f][1] = cx1 * C_DIM + cy0;  wgt[f][1] = s * (1.0f - t);
    off[f][2] = cx0 * C_DIM + cy1;  wgt[f][2] = (1.0f - s) * t;
    off[f][3] = cx1 * C_DIM + cy1;  wgt[f][3] = (1.0f - s) * (1.0f - t);
  }

  #pragma unroll
  for (int b = 0; b < B_DIM; ++b) {
    v8f acc = {};
    #pragma unroll
    for (int kk = 0; kk < 8; ++kk) {
      const int f = kk >> 1;                // compile-time: keeps off/wgt in VGPRs
      v2f bv;
      #pragma unroll
      for (int j = 0; j < 2; ++j) {
        // global channel k = 4*kk + 2*half + j ; c = k & 7 (f is k>>3 == kk>>1)
        const int c = ((kk & 1) << 2) + (half << 1) + j;
        const float* xp = x + (size_t)(b * C_IN + c) * PLANE;
        float v =        wgt[f][0] * xp[off[f][0]];
        v = fmaf(wgt[f][1], xp[off[f][1]], v);
        v = fmaf(wgt[f][2], xp[off[f][2]], v);
        v = fmaf(wgt[f][3], xp[off[f][3]], v);
        if (j == 0) bv.x = v; else bv.y = v;
      }
      // D[o][pix] += comb[o][k..k+3] * warped[k..k+3][pix]
      acc = __builtin_amdgcn_wmma_f32_16x16x4_f32(
          /*neg_a=*/false, a[kk], /*neg_b=*/false, bv,
          /*c_mod=*/(short)0, acc, /*reuse_a=*/false, /*reuse_b=*/false);
    }
    #pragma unroll
    for (int v = 0; v < 8; ++v) {
      const int o = v + 8 * half;
      out[(size_t)(b * O_DIM + o) * PLANE + p] = acc[v];
    }
  }
}

extern "C" void kernel_launch(void* const* d_in, const int* in_sizes, int n_in,
                              void* d_out, int out_size, void* d_ws, size_t ws_size,
                              hipStream_t stream) {
  const float* x    = (const float*)d_in[0];   // (8, 8, 450, 480) f32
  const float* flow = (const float*)d_in[1];   // (4, 450, 480, 2) f32
  const float* comb = (const float*)d_in[2];   // (16, 32) f32
  float* out = (float*)d_out;                  // (8, 16, 450, 480) f32

  const int tiles = PLANE / 16;                // 13500
  const int wavesPerBlock = 4;                 // 128 threads
  dim3 block(128);
  dim3 grid(tiles / wavesPerBlock);            // 3375, exact -> EXEC always full
  flow_combine_wmma<<<grid, block, 0, stream>>>(x, flow, comb, out);
}